// ModelNew_88716844466707
// MI455X (gfx1250) — compile-verified
//
#include <hip/hip_runtime.h>
#include <stdint.h>
#include <math.h>

// C[i,j] = A[i,i] * B[i,j]  -- pure HBM-streaming op (~512 MiB @ 23.3 TB/s ~= 23 us).
// CDNA5 path used: GLOBAL_LOAD_ASYNC_TO_LDS_B128 (ASYNCcnt) double-buffered staging
// + non-temporal B128 stores. WMMA is deliberately NOT used (AI = 0.125 FLOP/byte).

typedef float v4f __attribute__((ext_vector_type(4)));
typedef int   v4i __attribute__((ext_vector_type(4)));

#define THREADS 256
#define VEC 4
#define CHUNK (THREADS * VEC)   // 1024 floats = 4 KiB per chunk per block
#define ITERS 4
#define SPAN (CHUNK * ITERS)    // 4096 floats per block

#if defined(__HIP_DEVICE_COMPILE__) && __has_builtin(__builtin_amdgcn_global_load_async_to_lds_b128)
#define USE_ASYNC_BUILTIN 1
#else
#define USE_ASYNC_BUILTIN 0
#endif

#if defined(__HIP_DEVICE_COMPILE__) && __has_builtin(__builtin_amdgcn_s_wait_asynccnt)
#define WAIT_ASYNC(n) __builtin_amdgcn_s_wait_asynccnt(n)
#else
#define WAIT_ASYNC(n) asm volatile("s_wait_asynccnt %0" ::"i"(n) : "memory")
#endif

__device__ __forceinline__ void async_copy_b128(const float* gsrc, float* lptr) {
#if USE_ASYNC_BUILTIN
    // Param 0 is 'v4i __device__*' per clang diagnostic; HIP allows implicit
    // generic -> device/shared address-space conversion at the call.
    __builtin_amdgcn_global_load_async_to_lds_b128((v4i*)gsrc, (v4i*)lptr,
                                                   /*offset=*/0, /*cpol=*/0);
#else
    uint32_t loff = (uint32_t)(uintptr_t)lptr;   // low 32 bits of generic = LDS byte offset
    uint64_t ga = (uint64_t)(uintptr_t)gsrc;
    asm volatile("global_load_async_to_lds_b128 %0, %1, off"
                 :: "v"(loff), "v"(ga)
                 : "memory");
#endif
}

__global__ __launch_bounds__(THREADS) void diag_scale_async_kernel(
    const float* __restrict__ A, const float* __restrict__ B,
    float* __restrict__ C, int N)
{
    __shared__ __align__(16) float lbuf[2][CHUNK];

    const int row = blockIdx.y;                     // wave-uniform
    const size_t rowBase = (size_t)row * (size_t)N;
    const float d = A[rowBase + (size_t)row];       // uniform -> s_load

    const int tid = threadIdx.x;
    const size_t base = rowBase + (size_t)blockIdx.x * SPAN + (size_t)tid * VEC;
    const float* gsrc = B + base;
    float* gdst = C + base;

    float* lptr0 = &lbuf[0][tid * VEC];
    float* lptr1 = &lbuf[1][tid * VEC];

    // Prime the double buffer: two chunks in flight (ASYNCcnt = 2).
    async_copy_b128(gsrc, lptr0);
    async_copy_b128(gsrc + CHUNK, lptr1);

#pragma unroll
    for (int k = 0; k < ITERS; ++k) {
        // Async loads complete in order: asynccnt<=1 (resp. 0) => chunk k landed in LDS.
        if (k + 1 < ITERS) { WAIT_ASYNC(1); } else { WAIT_ASYNC(0); }

        const float* lsrc = (k & 1) ? lptr1 : lptr0;
        v4f v = *(const v4f*)lsrc;                  // ds_load_b128
        v4f r = v * d;

        // Re-arm this buffer for chunk k+2 (issued after the ds_load above).
        if (k + 2 < ITERS)
            async_copy_b128(gsrc + (size_t)(k + 2) * CHUNK, (k & 1) ? lptr1 : lptr0);

        // Write-once output: non-temporal B128 store, skip cache residency.
        __builtin_nontemporal_store(r, (v4f*)(gdst + (size_t)k * CHUNK));
    }
}

// Generic fallback for sizes not divisible by SPAN.
__global__ void diag_scale_fallback_kernel(const float* __restrict__ A,
                                           const float* __restrict__ B,
                                           float* __restrict__ C, int N)
{
    size_t total = (size_t)N * (size_t)N;
    size_t stride = (size_t)gridDim.x * blockDim.x;
    for (size_t idx = (size_t)blockIdx.x * blockDim.x + threadIdx.x; idx < total; idx += stride) {
        size_t row = idx / (size_t)N;
        C[idx] = A[row * (size_t)N + row] * B[idx];
    }
}

extern "C" void kernel_launch(void* const* d_in, const int* in_sizes, int n_in,
                              void* d_out, int out_size, void* d_ws, size_t ws_size,
                              hipStream_t stream) {
    (void)n_in; (void)out_size; (void)d_ws; (void)ws_size;
    const float* A = (const float*)d_in[0];
    const float* B = (const float*)d_in[1];
    float* C = (float*)d_out;

    const int N = (int)(sqrt((double)in_sizes[0]) + 0.5);  // A is N*N

    if (N > 0 && (N % SPAN) == 0) {
        dim3 grid((unsigned)(N / SPAN), (unsigned)N, 1);   // (2, 8192) for N=8192
        diag_scale_async_kernel<<<grid, THREADS, 0, stream>>>(A, B, C, N);
    } else {
        size_t total = (size_t)N * (size_t)N;
        unsigned blocks = (unsigned)((total + THREADS - 1) / THREADS);
        if (blocks > 65535u * 8u) blocks = 65535u * 8u;
        if (blocks == 0) blocks = 1;
        diag_scale_fallback_kernel<<<blocks, THREADS, 0, stream>>>(A, B, C, N);
    }
}